// ADFCell_7662221656362
// MI455X (gfx1250) — compile-verified
//
#include <hip/hip_runtime.h>

// ---------------- problem dims (fixed by reference) ----------------
#define B_SZ   64
#define L_SZ   4096
#define T_SZ   32          // taps == wave32 lanes
#define DEPTH  16          // async global->LDS pipeline depth (slots)
#define LRW    0.015625f   // 1/2^6
#define LRF    0.0078125f  // 1/2^7
#define GMAX   30.0f
#define EPSF   1e-9f

// ---------------- gfx1250 async-tensor helpers ----------------
typedef int v2i __attribute__((ext_vector_type(2)));
typedef __attribute__((address_space(1))) v2i g_v2i;   // global b64 chunk
typedef __attribute__((address_space(3))) v2i l_v2i;   // LDS b64 chunk

static __device__ __forceinline__ void async_copy_b64(const float* g, float* l) {
#if __has_builtin(__builtin_amdgcn_global_load_async_to_lds_b64)
  g_v2i* gp = (g_v2i*)(unsigned long long)(const void*)g;
  l_v2i* lp = (l_v2i*)(unsigned)(unsigned long long)(void*)l;
  __builtin_amdgcn_global_load_async_to_lds_b64(gp, lp, 0, 0);
#else
  unsigned loff = (unsigned)(unsigned long long)(void*)l;
  asm volatile("global_load_async_to_lds_b64 %0, %1, off"
               :: "v"(loff), "v"(g) : "memory");
#endif
}

static __device__ __forceinline__ void wait_async_deep() {
#if __has_builtin(__builtin_amdgcn_s_wait_asynccnt)
  __builtin_amdgcn_s_wait_asynccnt(2 * DEPTH - 2);
#else
  asm volatile("s_wait_asynccnt %0" :: "n"(2 * DEPTH - 2));
#endif
  asm volatile("" ::: "memory");
}
static __device__ __forceinline__ void wait_async_zero() {
#if __has_builtin(__builtin_amdgcn_s_wait_asynccnt)
  __builtin_amdgcn_s_wait_asynccnt(0);
#else
  asm volatile("s_wait_asynccnt 0");
#endif
  asm volatile("" ::: "memory");
}
static __device__ __forceinline__ void wait_ds_zero() {
  // slot recycle guard: ds reads of the slot (issued a full iteration ago)
  // must have landed in VGPRs before the async engine overwrites the slot
  asm volatile("s_wait_dscnt 0" ::: "memory");
}

// ---------------- fast transcendental helpers (single trans op) ----------------
static __device__ __forceinline__ float fast_rcp(float x) {
#if __has_builtin(__builtin_amdgcn_rcpf)
  return __builtin_amdgcn_rcpf(x);
#else
  return 1.0f / x;
#endif
}
static __device__ __forceinline__ float fast_rsq(float x) {
#if __has_builtin(__builtin_amdgcn_rsqf)
  return __builtin_amdgcn_rsqf(x);
#else
  return rsqrtf(x);
#endif
}

// ---------------- 32-lane all-reduce, VALU-pipe only ----------------
#if __has_builtin(__builtin_amdgcn_update_dpp) && __has_builtin(__builtin_amdgcn_permlanex16)
template <int CTRL>
static __device__ __forceinline__ float add_dpp(float v) {
  const int s = __builtin_amdgcn_update_dpp(0, __float_as_int(v), CTRL, 0xF, 0xF, true);
  return v + __int_as_float(s);
}
static __device__ __forceinline__ float wave_allsum(float v) {
  v = add_dpp<0x0B1>(v);   // quad_perm [1,0,3,2]  : xor 1
  v = add_dpp<0x04E>(v);   // quad_perm [2,3,0,1]  : xor 2
  v = add_dpp<0x164>(v);   // row_xmask:4          : xor 4
  v = add_dpp<0x168>(v);   // row_xmask:8          : xor 8
  const int x = __builtin_amdgcn_permlanex16(__float_as_int(v), __float_as_int(v),
                                             0x76543210, (int)0xfedcba98, false, false);
  return v + __int_as_float(x);   // xor 16: swap 16-lane halves
}
#else
static __device__ __forceinline__ float wave_allsum(float v) {
#pragma unroll
  for (int off = 16; off > 0; off >>= 1)
    v += __shfl_xor(v, off, 32);
  return v;
}
#endif

// ---------------- one ADF recurrence step (state in registers) ----------------
static __device__ __forceinline__ void adf_step(
    float (&wr)[2][2], float (&wi)[2][2], float (&fr)[2], float (&fi)[2],
    float ur0, float ur1, float ui0, float ui1,
    float2 xr, float2 xi, int t, float* outp)
{
  // per-lane partial matvec v = W u and |u|^2
  float pvr0 = wr[0][0]*ur0 - wi[0][0]*ui0 + wr[0][1]*ur1 - wi[0][1]*ui1;
  float pvi0 = wr[0][0]*ui0 + wi[0][0]*ur0 + wr[0][1]*ui1 + wi[0][1]*ur1;
  float pvr1 = wr[1][0]*ur0 - wi[1][0]*ui0 + wr[1][1]*ur1 - wi[1][1]*ui1;
  float pvi1 = wr[1][0]*ui0 + wi[1][0]*ur0 + wr[1][1]*ui1 + wi[1][1]*ur1;
  float pue  = ur0*ur0 + ui0*ui0 + ur1*ur1 + ui1*ui1;

  float vr[2], vi[2];
  vr[0] = wave_allsum(pvr0);  vi[0] = wave_allsum(pvi0);
  vr[1] = wave_allsum(pvr1);  vi[1] = wave_allsum(pvi1);
  const float inv_ue = fast_rcp(wave_allsum(pue) + EPSF);

  const float dr[2] = {xr.x, xr.y};
  const float di[2] = {xi.x, xi.y};

  float kr[2], ki[2], ewr[2], ewi[2], efr[2], efi[2];
#pragma unroll
  for (int m = 0; m < 2; ++m) {
    // k = v * f
    kr[m] = vr[m]*fr[m] - vi[m]*fi[m];
    ki[m] = vr[m]*fi[m] + vi[m]*fr[m];
    // psi_hat = |f|/f = conj(f) * rsq(|f|^2)
    const float rs  = fast_rsq(fr[m]*fr[m] + fi[m]*fi[m]);
    const float pr  = fr[m] * rs;
    const float pi_ = -fi[m] * rs;
    // e_w = d*psi - v ;  e_f = d - k
    ewr[m] = (dr[m]*pr  - di[m]*pi_) - vr[m];
    ewi[m] = (dr[m]*pi_ + di[m]*pr ) - vi[m];
    efr[m] = dr[m] - kr[m];
    efi[m] = di[m] - ki[m];
  }

  const float inv_ve =
      fast_rcp(vr[0]*vr[0] + vi[0]*vi[0] + vr[1]*vr[1] + vi[1]*vi[1] + EPSF);
#pragma unroll
  for (int m = 0; m < 2; ++m) {
    // gf = -(1/ve) * e_f * conj(v), magnitude-clipped at GMAX (no sqrt/div)
    float gfr = -inv_ve * (efr[m]*vr[m] + efi[m]*vi[m]);
    float gfi = -inv_ve * (efi[m]*vr[m] - efr[m]*vi[m]);
    const float ag2 = gfr*gfr + gfi*gfi;
    const float sc  = (ag2 > GMAX*GMAX) ? GMAX * fast_rsq(ag2) : 1.0f;
    fr[m] -= LRF * gfr * sc;
    fi[m] -= LRF * gfi * sc;
  }

  // w += LRW * (1/ue) * e_w[i] * conj(u[j])   (== w - LRW*gw)
  const float cw = LRW * inv_ue;
  const float ar[2] = {cw*ewr[0], cw*ewr[1]};
  const float ai[2] = {cw*ewi[0], cw*ewi[1]};
  const float urj[2] = {ur0, ur1};
  const float uij[2] = {ui0, ui1};
#pragma unroll
  for (int i = 0; i < 2; ++i)
#pragma unroll
    for (int j = 0; j < 2; ++j) {
      wr[i][j] += ar[i]*urj[j] + ai[i]*uij[j];
      wi[i][j] += ai[i]*urj[j] - ar[i]*uij[j];
    }

  // output k -> [B,L,Nm,2]
  if (t == 0) {
    float4 o;
    o.x = kr[0]; o.y = ki[0]; o.z = kr[1]; o.w = ki[1];
    *(float4*)outp = o;
  }
}

// ---------------- kernel: one wave per batch chain ----------------
__global__ __launch_bounds__(32, 1)
void adf_scan_kernel(const float* __restrict__ u_r, const float* __restrict__ u_i,
                     const float* __restrict__ x_r, const float* __restrict__ x_i,
                     const float* __restrict__ w0_r, const float* __restrict__ w0_i,
                     const float* __restrict__ f0_r, const float* __restrict__ f0_i,
                     float* __restrict__ out)
{
  __shared__ float lds_u[DEPTH * 128];   // per slot: 64 floats u_r | 64 floats u_i

  const int b = blockIdx.x;       // batch chain
  const int t = threadIdx.x;      // tap lane (0..31)

  // ---- recurrent state in registers ----
  float wr[2][2], wi[2][2];
#pragma unroll
  for (int i = 0; i < 2; ++i)
#pragma unroll
    for (int j = 0; j < 2; ++j) {
      const int idx = ((b * 2 + i) * 2 + j) * T_SZ + t;   // [B,Nm,Nm,T]
      wr[i][j] = w0_r[idx];
      wi[i][j] = w0_i[idx];
    }
  float fr[2], fi[2];
  fr[0] = f0_r[b * 2 + 0]; fr[1] = f0_r[b * 2 + 1];
  fi[0] = f0_i[b * 2 + 0]; fi[1] = f0_i[b * 2 + 1];

  const size_t ub = (size_t)b * L_SZ * 64;   // u arrays: [B,L,T,Nm] floats
  const size_t xb = (size_t)b * L_SZ * 2;    // x arrays: [B,L,Nm]
  float* const outb = out + (size_t)b * L_SZ * 4;

  // ---- prologue: fill DEPTH-slot async pipeline (ASYNCcnt-tracked) ----
#pragma unroll
  for (int p = 0; p < DEPTH; ++p) {
    async_copy_b64(u_r + ub + (size_t)p * 64 + t * 2, &lds_u[p * 128 + t * 2]);
    async_copy_b64(u_i + ub + (size_t)p * 64 + t * 2, &lds_u[p * 128 + 64 + t * 2]);
  }

  // register double-buffer for u: read slot 0 now, slot l+1 during iter l
  wait_async_deep();                               // step 0 landed
  float2 ur_c = *(const float2*)&lds_u[t * 2];
  float2 ui_c = *(const float2*)&lds_u[64 + t * 2];

  float2 xr_c = *(const float2*)(x_r + xb);
  float2 xi_c = *(const float2*)(x_i + xb);

  // ---- main loop: branch-free, always reissues ----
#pragma unroll 2
  for (int l = 0; l < L_SZ - DEPTH; ++l) {
    // next-step x loads (consumed at end of iteration)
    const float2 xr_n = *(const float2*)(x_r + xb + (size_t)(l + 1) * 2);
    const float2 xi_n = *(const float2*)(x_i + xb + (size_t)(l + 1) * 2);

    // recycle slot l for step l+DEPTH (its ds reads are an iteration old)
    wait_ds_zero();
    const int slot = l & (DEPTH - 1);
    const size_t s = (size_t)(l + DEPTH) * 64 + t * 2;
    async_copy_b64(u_r + ub + s, &lds_u[slot * 128 + t * 2]);
    async_copy_b64(u_i + ub + s, &lds_u[slot * 128 + 64 + t * 2]);

    // prefetch step l+1 from LDS into the register buffer (overlaps compute)
    wait_async_deep();                             // step l+1 landed
    const int nslot = (l + 1) & (DEPTH - 1);
    const float2 ur_n = *(const float2*)&lds_u[nslot * 128 + t * 2];
    const float2 ui_n = *(const float2*)&lds_u[nslot * 128 + 64 + t * 2];

    adf_step(wr, wi, fr, fi, ur_c.x, ur_c.y, ui_c.x, ui_c.y,
             xr_c, xi_c, t, outb + (size_t)l * 4);

    ur_c = ur_n; ui_c = ui_n;
    xr_c = xr_n; xi_c = xi_n;
  }

  // ---- tail: everything already issued; drain ----
  wait_async_zero();
  for (int l = L_SZ - DEPTH; l < L_SZ; ++l) {
    const int ln = (l + 1 < L_SZ) ? l + 1 : L_SZ - 1;
    const float2 xr_n = *(const float2*)(x_r + xb + (size_t)ln * 2);
    const float2 xi_n = *(const float2*)(x_i + xb + (size_t)ln * 2);

    const int slot = l & (DEPTH - 1);
    const float2 ur2 = *(const float2*)&lds_u[slot * 128 + t * 2];
    const float2 ui2 = *(const float2*)&lds_u[slot * 128 + 64 + t * 2];

    adf_step(wr, wi, fr, fi, ur2.x, ur2.y, ui2.x, ui2.y,
             xr_c, xi_c, t, outb + (size_t)l * 4);

    xr_c = xr_n; xi_c = xi_n;
  }
}

// ---------------- host launch ----------------
extern "C" void kernel_launch(void* const* d_in, const int* in_sizes, int n_in,
                              void* d_out, int out_size, void* d_ws, size_t ws_size,
                              hipStream_t stream) {
  (void)in_sizes; (void)n_in; (void)out_size; (void)d_ws; (void)ws_size;
  const float* u_r  = (const float*)d_in[0];
  const float* u_i  = (const float*)d_in[1];
  const float* x_r  = (const float*)d_in[2];
  const float* x_i  = (const float*)d_in[3];
  const float* w0_r = (const float*)d_in[4];
  const float* w0_i = (const float*)d_in[5];
  const float* f0_r = (const float*)d_in[6];
  const float* f0_i = (const float*)d_in[7];
  float* out = (float*)d_out;

  adf_scan_kernel<<<dim3(B_SZ), dim3(T_SZ), 0, stream>>>(
      u_r, u_i, x_r, x_i, w0_r, w0_i, f0_r, f0_i, out);
}